// feature_selection_GNN_2473901162529
// MI455X (gfx1250) — compile-verified
//
#include <hip/hip_runtime.h>
#include <math.h>

typedef __attribute__((ext_vector_type(2))) float v2f;
typedef __attribute__((ext_vector_type(8))) float v8f;

// ---- problem constants (match reference) ----
static constexpr int G_    = 256;
static constexpr int NPG_  = 1000;
static constexpr int NNODE = G_ * NPG_;   // 256000
static constexpr int FIN_  = 116;
static constexpr int C_    = 50;
static constexpr int K1_   = 500;
static constexpr int K2_   = 250;
static constexpr int NCLS_ = 4;

// =====================================================================
// WMMA f32 GEMM:  C[M,N] = A[M,K] @ B[K,N]   (fp32, V_WMMA_F32_16X16X4_F32)
//  - K is a compile-time constant; loop fully unrolled, no predication.
//  - B staged in LDS, TRANSPOSED and zero-padded to [NP][KP]; pad rows are
//    0.0 so A may over-read up to 2 floats past its row end (buffer slack
//    is guaranteed by the workspace allocator).
//  - M must be a multiple of 16. One wave per 16x16 tile, 8 waves/block.
// =====================================================================
template <int K, int NTN>
__global__ void wmma_gemm_f32(const float* __restrict__ A,
                              const float* __restrict__ B,
                              float* __restrict__ C,
                              int M, int N) {
  constexpr int KP = (K + 3) & ~3;   // K padded to multiple of 4
  constexpr int NP = NTN * 16;       // N padded to multiple of 16
  __shared__ float Bs[NP * KP];      // transposed: Bs[n][k]

  // cooperative zero-padded transpose load of B into LDS
  for (int i = threadIdx.x; i < NP * KP; i += blockDim.x) {
    const int n = i / KP, k = i % KP;
    Bs[i] = (k < K && n < N) ? B[k * N + n] : 0.0f;
  }
  __syncthreads();

  const int wave = threadIdx.x >> 5;
  const int lane = threadIdx.x & 31;
  const long tile = (long)blockIdx.x * (blockDim.x >> 5) + wave;
  const long nTiles = (long)(M >> 4) * NTN;
  if (tile >= nTiles) return;        // wave-uniform: EXEC all-ones at WMMA

  const int mt = (int)(tile / NTN);
  const int nt = (int)(tile % NTN);
  const int m0 = mt << 4, n0 = nt << 4;
  const int mm = lane & 15;          // row (A) / col (B) within tile
  const int half = lane >> 4;        // K sub-pair selector per ISA layout

  const float* arow = A + (long)(m0 + mm) * K + half * 2;
  const float* bcol = Bs + (n0 + mm) * KP + half * 2;

  v8f acc = {};
#pragma unroll
  for (int k = 0; k < KP; k += 4) {
    // (m0+mm)*K is even and k+half*2 is even for K in {116,50} -> 8B aligned
    const v2f a = *(const v2f*)(arow + k);
    const v2f b = *(const v2f*)(bcol + k);
    acc = __builtin_amdgcn_wmma_f32_16x16x4_f32(
        /*neg_a=*/false, a, /*neg_b=*/false, b,
        /*c_mod=*/(short)0, acc, /*reuse_a=*/false, /*reuse_b=*/false);
  }

  const int ncol = n0 + mm;
  const bool nok = ncol < N;
#pragma unroll
  for (int r = 0; r < 8; ++r) {
    const int row = m0 + r + half * 8;   // 16x16 f32 C/D VGPR layout
    if (nok) C[(long)row * N + ncol] = acc[r];
  }
}

// =====================================================================
// degree accumulation (ew == 1), optionally via relabel map
// =====================================================================
__global__ void deg1_kernel(const int* __restrict__ dst, float* __restrict__ deg, int E) {
  int e = blockIdx.x * blockDim.x + threadIdx.x;
  if (e < E) atomicAdd(&deg[dst[e]], 1.0f);
}

__global__ void deg2_kernel(const int* __restrict__ src, const int* __restrict__ dst,
                            const int* __restrict__ inv, float* __restrict__ deg, int E) {
  int e = blockIdx.x * blockDim.x + threadIdx.x;
  if (e >= E) return;
  int ns = inv[src[e]], nd = inv[dst[e]];
  if (ns >= 0 && nd >= 0) atomicAdd(&deg[nd], 1.0f);
}

__global__ void rsqrt_kernel(float* __restrict__ d, int n) {
  int i = blockIdx.x * blockDim.x + threadIdx.x;
  if (i < n) d[i] = rsqrtf(d[i] + 1.0f);   // +1 self loop
}

// =====================================================================
// edge aggregation: agg[dst] += h[src] * dis[src]*dis[dst]  (warp/edge)
// =====================================================================
__global__ void agg1_kernel(const float* __restrict__ h,
                            const int* __restrict__ src, const int* __restrict__ dst,
                            const float* __restrict__ dis, float* __restrict__ agg,
                            int E, int C) {
  int warp = (blockIdx.x * blockDim.x + threadIdx.x) >> 5;
  int lane = threadIdx.x & 31;
  if (warp >= E) return;
  int s = src[warp], d = dst[warp];
  float coef = dis[s] * dis[d];
  const float* hrow = h + (long)s * C;
  float* arow = agg + (long)d * C;
  for (int c = lane; c < C; c += 32)
    atomicAdd(&arow[c], hrow[c] * coef);
}

__global__ void agg2_kernel(const float* __restrict__ h,
                            const int* __restrict__ src, const int* __restrict__ dst,
                            const int* __restrict__ inv,
                            const float* __restrict__ dis, float* __restrict__ agg,
                            int E, int C) {
  int warp = (blockIdx.x * blockDim.x + threadIdx.x) >> 5;
  int lane = threadIdx.x & 31;
  if (warp >= E) return;
  int ns = inv[src[warp]], nd = inv[dst[warp]];
  if (ns < 0 || nd < 0) return;            // ew == 0: contributes nothing
  float coef = dis[ns] * dis[nd];
  const float* hrow = h + (long)ns * C;
  float* arow = agg + (long)nd * C;
  for (int c = lane; c < C; c += 32)
    atomicAdd(&arow[c], hrow[c] * coef);
}

// h_out = agg + h_lin * dis^2 + bias      (in-place into agg)
__global__ void combine_kernel(float* __restrict__ agg, const float* __restrict__ hlin,
                               const float* __restrict__ dis, const float* __restrict__ bias,
                               long nnode, int C) {
  long idx = (long)blockIdx.x * blockDim.x + threadIdx.x;
  if (idx >= nnode * C) return;
  int r = (int)(idx / C), c = (int)(idx % C);
  float di = dis[r];
  agg[idx] = agg[idx] + hlin[idx] * di * di + bias[c];
}

// =====================================================================
// ||p|| (single small block)
// =====================================================================
__global__ void norm_kernel(const float* __restrict__ p, int C, float* __restrict__ out) {
  __shared__ float red[64];
  float s = 0.0f;
  for (int i = threadIdx.x; i < C; i += 64) s += p[i] * p[i];
  red[threadIdx.x] = s;
  __syncthreads();
  for (int off = 32; off > 0; off >>= 1) {
    if (threadIdx.x < off) red[threadIdx.x] += red[threadIdx.x + off];
    __syncthreads();
  }
  if (threadIdx.x == 0) *out = sqrtf(red[0]);
}

// =====================================================================
// per-graph top-k: score = tanh(h.p/||p||), LDS bitonic sort descending
// =====================================================================
template <int NS, int NPER, int KK>
__global__ void topk_kernel(const float* __restrict__ h, const float* __restrict__ p,
                            const float* __restrict__ pnorm, int C,
                            float* __restrict__ vals, int* __restrict__ perm) {
  __shared__ float sv[NS];
  __shared__ int   si[NS];
  const int g = blockIdx.x;
  const int tid = threadIdx.x;
  const float pn = *pnorm;

  for (int i = tid; i < NS; i += blockDim.x) {
    float s = -1.0e30f;
    if (i < NPER) {
      const float* row = h + (long)(g * NPER + i) * C;
      float d = 0.0f;
      for (int c = 0; c < C; ++c) d += row[c] * p[c];
      s = tanhf(d / pn);
    }
    sv[i] = s;
    si[i] = i;
  }
  __syncthreads();

  for (int k = 2; k <= NS; k <<= 1) {
    for (int j = k >> 1; j > 0; j >>= 1) {
      for (int i = tid; i < NS; i += blockDim.x) {
        int ixj = i ^ j;
        if (ixj > i) {
          bool descend = ((i & k) == 0);
          float vi = sv[i], vj = sv[ixj];
          bool sw = descend ? (vi < vj) : (vi > vj);
          if (sw) {
            sv[i] = vj; sv[ixj] = vi;
            int t = si[i]; si[i] = si[ixj]; si[ixj] = t;
          }
        }
      }
      __syncthreads();
    }
  }

  for (int t = tid; t < KK; t += blockDim.x) {
    vals[g * KK + t] = sv[t];
    perm[g * KK + t] = g * NPER + si[t];
  }
}

// x_out[r,:] = h[perm[r],:] * vals[r]
__global__ void gather_scale_kernel(const float* __restrict__ h, const int* __restrict__ perm,
                                    const float* __restrict__ vals, float* __restrict__ xo,
                                    long rows, int C) {
  long idx = (long)blockIdx.x * blockDim.x + threadIdx.x;
  if (idx >= rows * C) return;
  int r = (int)(idx / C), c = (int)(idx % C);
  xo[idx] = h[(long)perm[r] * C + c] * vals[r];
}

__global__ void scatter_inv_kernel(const int* __restrict__ perm, int* __restrict__ inv, int rows) {
  int i = blockIdx.x * blockDim.x + threadIdx.x;
  if (i < rows) inv[perm[i]] = i;
}

// =====================================================================
// readout: [max | mean] over k contiguous nodes per graph
// =====================================================================
__global__ void readout_kernel(const float* __restrict__ x, int k, int C,
                               float* __restrict__ xx, int ldxx, int colOff) {
  int g = blockIdx.x;
  int t = threadIdx.x;
  if (t >= C) return;
  const float* base = x + (long)g * k * C + t;
  float mx = -1.0e30f, sm = 0.0f;
  for (int i = 0; i < k; ++i) {
    float v = base[(long)i * C];
    mx = fmaxf(mx, v);
    sm += v;
  }
  xx[(long)g * ldxx + colOff + t]     = mx;
  xx[(long)g * ldxx + colOff + C + t] = sm / (float)k;
}

// =====================================================================
// head: out = log_softmax(xx @ Wf + bf)   — one thread per graph
// =====================================================================
__global__ void head_kernel(const float* __restrict__ xx, const float* __restrict__ Wf,
                            const float* __restrict__ bf, float* __restrict__ out,
                            int nG, int Din, int ncls) {
  int g = blockIdx.x * blockDim.x + threadIdx.x;
  if (g >= nG) return;
  float logits[NCLS_];
  for (int c = 0; c < ncls; ++c) {
    float s = bf[c];
    const float* row = xx + (long)g * Din;
    for (int i = 0; i < Din; ++i) s += row[i] * Wf[(long)i * ncls + c];
    logits[c] = s;
  }
  float mx = logits[0];
  for (int c = 1; c < ncls; ++c) mx = fmaxf(mx, logits[c]);
  float se = 0.0f;
  for (int c = 0; c < ncls; ++c) se += expf(logits[c] - mx);
  float lse = mx + logf(se);
  for (int c = 0; c < ncls; ++c) out[(long)g * ncls + c] = logits[c] - lse;
}

// =====================================================================
// launch
// =====================================================================
extern "C" void kernel_launch(void* const* d_in, const int* in_sizes, int n_in,
                              void* d_out, int out_size, void* d_ws, size_t ws_size,
                              hipStream_t stream) {
  const float* x  = (const float*)d_in[0];
  const int*   ei = (const int*)d_in[1];
  const float* W1 = (const float*)d_in[2];
  const float* b1 = (const float*)d_in[3];
  const float* W2 = (const float*)d_in[4];
  const float* b2 = (const float*)d_in[5];
  const float* p1 = (const float*)d_in[6];
  const float* p2 = (const float*)d_in[7];
  const float* Wf = (const float*)d_in[8];
  const float* bf = (const float*)d_in[9];

  const int E = in_sizes[1] / 2;
  const int* src = ei;
  const int* dst = ei + E;

  // ---- workspace layout (256B-aligned blocks -> >=8B slack after each) ----
  char* ws = (char*)d_ws;
  size_t off = 0;
  auto alloc = [&](size_t bytes) -> void* {
    void* p = ws + off;
    off += (bytes + 255) & ~(size_t)255;
    return p;
  };
  float* h1lin = (float*)alloc(sizeof(float) * (size_t)NNODE * C_);
  float* h1    = (float*)alloc(sizeof(float) * (size_t)NNODE * C_);   // agg1 -> h1
  float* dis1  = (float*)alloc(sizeof(float) * (size_t)NNODE);        // deg1 -> dis1
  int*   inv   = (int*)  alloc(sizeof(int)   * (size_t)NNODE);
  float* vals1 = (float*)alloc(sizeof(float) * (size_t)G_ * K1_);
  int*   perm1 = (int*)  alloc(sizeof(int)   * (size_t)G_ * K1_);
  float* x1    = (float*)alloc(sizeof(float) * (size_t)G_ * K1_ * C_);
  float* h2lin = (float*)alloc(sizeof(float) * (size_t)G_ * K1_ * C_);
  float* h2    = (float*)alloc(sizeof(float) * (size_t)G_ * K1_ * C_); // agg2 -> h2
  float* dis2  = (float*)alloc(sizeof(float) * (size_t)G_ * K1_);      // deg2 -> dis2
  float* vals2 = (float*)alloc(sizeof(float) * (size_t)G_ * K2_);
  int*   perm2 = (int*)  alloc(sizeof(int)   * (size_t)G_ * K2_);
  float* x2    = (float*)alloc(sizeof(float) * (size_t)G_ * K2_ * C_);
  float* xx    = (float*)alloc(sizeof(float) * (size_t)G_ * 4 * C_);
  float* pn1   = (float*)alloc(sizeof(float));
  float* pn2   = (float*)alloc(sizeof(float));

  // ---- init accumulators (graph-capture safe) ----
  hipMemsetAsync(h1,   0,    sizeof(float) * (size_t)NNODE * C_, stream);
  hipMemsetAsync(dis1, 0,    sizeof(float) * (size_t)NNODE, stream);
  hipMemsetAsync(h2,   0,    sizeof(float) * (size_t)G_ * K1_ * C_, stream);
  hipMemsetAsync(dis2, 0,    sizeof(float) * (size_t)G_ * K1_, stream);
  hipMemsetAsync(inv,  0xFF, sizeof(int)   * (size_t)NNODE, stream);  // -1

  norm_kernel<<<1, 64, 0, stream>>>(p1, C_, pn1);
  norm_kernel<<<1, 64, 0, stream>>>(p2, C_, pn2);

  // ---- GCN layer 1 :  h1lin = x @ W1   (M=256000, K=116, N=50) ----
  {
    const int M = NNODE, N = C_;
    long tiles = (long)(M / 16) * 4;
    unsigned blocks = (unsigned)((tiles + 7) / 8);
    wmma_gemm_f32<FIN_, 4><<<blocks, 256, 0, stream>>>(x, W1, h1lin, M, N);
  }
  deg1_kernel<<<(E + 255) / 256, 256, 0, stream>>>(dst, dis1, E);
  rsqrt_kernel<<<(NNODE + 255) / 256, 256, 0, stream>>>(dis1, NNODE);
  agg1_kernel<<<(unsigned)((E + 7) / 8), 256, 0, stream>>>(h1lin, src, dst, dis1, h1, E, C_);
  {
    long tot = (long)NNODE * C_;
    combine_kernel<<<(unsigned)((tot + 255) / 256), 256, 0, stream>>>(h1, h1lin, dis1, b1, NNODE, C_);
  }

  // ---- top-k pool 1 (1000 -> 500 per graph) ----
  topk_kernel<1024, NPG_, K1_><<<G_, 256, 0, stream>>>(h1, p1, pn1, C_, vals1, perm1);
  {
    long rows = (long)G_ * K1_;
    gather_scale_kernel<<<(unsigned)((rows * C_ + 255) / 256), 256, 0, stream>>>(
        h1, perm1, vals1, x1, rows, C_);
    scatter_inv_kernel<<<(unsigned)((rows + 255) / 256), 256, 0, stream>>>(perm1, inv, (int)rows);
  }

  // ---- GCN layer 2 :  h2lin = x1 @ W2  (M=128000, K=50, N=50) ----
  {
    const int M = G_ * K1_, N = C_;
    long tiles = (long)(M / 16) * 4;
    unsigned blocks = (unsigned)((tiles + 7) / 8);
    wmma_gemm_f32<C_, 4><<<blocks, 256, 0, stream>>>(x1, W2, h2lin, M, N);
  }
  deg2_kernel<<<(E + 255) / 256, 256, 0, stream>>>(src, dst, inv, dis2, E);
  rsqrt_kernel<<<(G_ * K1_ + 255) / 256, 256, 0, stream>>>(dis2, G_ * K1_);
  agg2_kernel<<<(unsigned)((E + 7) / 8), 256, 0, stream>>>(h2lin, src, dst, inv, dis2, h2, E, C_);
  {
    long tot = (long)G_ * K1_ * C_;
    combine_kernel<<<(unsigned)((tot + 255) / 256), 256, 0, stream>>>(h2, h2lin, dis2, b2, (long)G_ * K1_, C_);
  }

  // ---- readout 1 : xx[:, 0:100] = [gmp1 | gap1] ----
  readout_kernel<<<G_, 64, 0, stream>>>(x1, K1_, C_, xx, 4 * C_, 0);

  // ---- top-k pool 2 (500 -> 250 per graph) ----
  topk_kernel<512, K1_, K2_><<<G_, 256, 0, stream>>>(h2, p2, pn2, C_, vals2, perm2);
  {
    long rows = (long)G_ * K2_;
    gather_scale_kernel<<<(unsigned)((rows * C_ + 255) / 256), 256, 0, stream>>>(
        h2, perm2, vals2, x2, rows, C_);
  }

  // ---- readout 2 : xx[:, 100:200] = [gmp2 | gap2] ----
  readout_kernel<<<G_, 64, 0, stream>>>(x2, K2_, C_, xx, 4 * C_, 2 * C_);

  // ---- head + log_softmax ----
  head_kernel<<<1, G_, 0, stream>>>(xx, Wf, bf, (float*)d_out, G_, 4 * C_, NCLS_);
}